// memAttention_72267119722884
// MI455X (gfx1250) — compile-verified
//
#include <hip/hip_runtime.h>
#include <hip/hip_bf16.h>
#include <stdint.h>

// Problem constants (from reference setup_inputs)
#define NN 4096
#define SS 200
#define XX 64
#define MM 364
#define OO 64

typedef float v2f __attribute__((ext_vector_type(2)));
typedef float v8f __attribute__((ext_vector_type(8)));

// ---------------------------------------------------------------------------
// Kernel 1 & 3: C = relu(A @ B + bias)  via V_WMMA_F32_16X16X4_F32
//   A: (nrow x K) row-major lda, B: (K x ncol) row-major ldb, C: row-major ldc
//   nrow multiple of 16; K multiple of 4; ncol arbitrary.
//   One wave32 per 16x16 output tile.
//   Column padding trick: output column c depends only on B column c, so for
//   padded columns we just CLAMP the load address (stay in-bounds) and never
//   store — no zero-masking, no EXEC divergence in the WMMA loop.
// ---------------------------------------------------------------------------
__global__ __launch_bounds__(128) void gemm_bias_relu_wmma(
    const float* __restrict__ A, const float* __restrict__ B,
    const float* __restrict__ bias, float* __restrict__ C,
    int nrow, int K, int ncol, int lda, int ldb, int ldc)
{
    const int lane  = threadIdx.x & 31;
    const int wtile = blockIdx.x * (blockDim.x >> 5) + (threadIdx.x >> 5);
    const int colTiles = (ncol + 15) >> 4;
    const int rowTiles = nrow >> 4;
    if (wtile >= rowTiles * colTiles) return;   // wave-uniform exit: EXEC all-1 for WMMA

    const int r0 = (wtile / colTiles) << 4;
    const int c0 = (wtile % colTiles) << 4;
    const int hi = lane >> 4;       // lane half: selects K pair (A/B) and row half (C)
    const int lo = lane & 15;
    const int row  = r0 + lo;                         // A row for this lane
    const int col  = c0 + lo;                         // logical B/C column
    const bool colOK = (col < ncol);
    const int ccol = colOK ? col : (ncol - 1);        // clamped (in-bounds) column

    const float* Arow = A + (size_t)row * lda;
    const float* Bcol = B + ccol;

    v8f acc = {0.f, 0.f, 0.f, 0.f, 0.f, 0.f, 0.f, 0.f};

    #pragma unroll 4
    for (int k = 0; k < K; k += 4) {
        const int kk = k + hi * 2;          // lanes 0-15: k,k+1 ; lanes 16-31: k+2,k+3
        v2f a, b;
        a.x = Arow[kk];
        a.y = Arow[kk + 1];
        b.x = Bcol[(size_t)kk * ldb];       // unconditional, address clamped
        b.y = Bcol[(size_t)(kk + 1) * ldb];
        acc = __builtin_amdgcn_wmma_f32_16x16x4_f32(
            /*neg_a=*/false, a, /*neg_b=*/false, b,
            /*c_mod=*/(short)0, acc, /*reuse_a=*/false, /*reuse_b=*/false);
    }

    if (colOK) {
        const float bv = bias[col];
        #pragma unroll
        for (int i = 0; i < 8; ++i) {
            const int r = r0 + i + hi * 8;  // C VGPR i: rows i (lanes 0-15) / i+8 (lanes 16-31)
            float v = acc[i] + bv;
            C[(size_t)r * ldc + col] = v > 0.f ? v : 0.f;
        }
    }
}

// ---------------------------------------------------------------------------
// Reductions (deterministic: fixed shuffle tree, no float atomics)
// ---------------------------------------------------------------------------
__device__ __forceinline__ float wave_sum(float v) {
    #pragma unroll
    for (int o = 16; o > 0; o >>= 1) v += __shfl_xor(v, o, 32);
    return v;
}
__device__ __forceinline__ float wave_max(float v) {
    #pragma unroll
    for (int o = 16; o > 0; o >>= 1) v = fmaxf(v, __shfl_xor(v, o, 32));
    return v;
}
__device__ __forceinline__ float block_sum(float v, float* sred) {
    const int tid = threadIdx.x, wid = tid >> 5, lid = tid & 31;
    v = wave_sum(v);
    __syncthreads();
    if (lid == 0) sred[wid] = v;
    __syncthreads();
    if (wid == 0) {
        float x = (lid < (int)(blockDim.x >> 5)) ? sred[lid] : 0.f;
        x = wave_sum(x);
        if (lid == 0) sred[16] = x;
    }
    __syncthreads();
    return sred[16];
}
__device__ __forceinline__ float block_max(float v, float* sred) {
    const int tid = threadIdx.x, wid = tid >> 5, lid = tid & 31;
    v = wave_max(v);
    __syncthreads();
    if (lid == 0) sred[wid] = v;
    __syncthreads();
    if (wid == 0) {
        float x = (lid < (int)(blockDim.x >> 5)) ? sred[lid] : -3.402823466e38f;
        x = wave_max(x);
        if (lid == 0) sred[16] = x;
    }
    __syncthreads();
    return sred[16];
}

// ---------------------------------------------------------------------------
// Kernel 2: one workgroup per n.
//   mem[n] (291,200 B) -> LDS via async global->LDS (single HBM pass, only
//   possible thanks to CDNA5's 320KB-per-WGP LDS),
//   sim = cos-sim(mem[n,s], h[n]) ; softmax over s ; pooled = mem^T w / S ;
//   g = pooled * h  -> workspace.
// ---------------------------------------------------------------------------
#define ATTN_BS 512

__global__ __launch_bounds__(ATTN_BS) void attn_kernel(
    const float* __restrict__ mem, const float* __restrict__ h_ws,
    float* __restrict__ g_ws)
{
    __shared__ float smem[SS * MM];   // 291,200 B
    __shared__ float sh[MM];
    __shared__ float ssim[SS];
    __shared__ float sred[17];

    const int n   = blockIdx.x;
    const int tid = threadIdx.x;
    const int bs  = ATTN_BS;
    const float* memn = mem + (size_t)n * SS * MM;

    // ---- async DMA of mem[n] into LDS (b128 per lane) ----
    const unsigned lds_base = (unsigned)(uintptr_t)(void*)smem;  // low 32 bits of flat LDS addr = LDS offset
    const int nvec = (SS * MM) / 4;                              // 18200 x 16B
    for (int i = tid; i < nvec; i += bs) {
        const unsigned lo = lds_base + (unsigned)i * 16u;
        const uint64_t gp = (uint64_t)(uintptr_t)(memn + (size_t)i * 4);
        asm volatile("global_load_async_to_lds_b128 %0, %1, off"
                     :: "v"(lo), "v"(gp) : "memory");
    }

    // ---- overlap: load h[n], compute ||h|| while DMA is in flight ----
    float acc = 0.f;
    for (int m = tid; m < MM; m += bs) {
        const float hv = h_ws[(size_t)n * MM + m];
        sh[m] = hv;
        acc += hv * hv;
    }
    const float hnorm = fmaxf(sqrtf(block_sum(acc, sred)), 1e-6f);

    asm volatile("s_wait_asynccnt 0" ::: "memory");
    __syncthreads();

    // ---- sim[s] = dot(mem[s], h) / (max(||mem[s]||,eps) * max(||h||,eps)) ----
    const int wid = tid >> 5, lane = tid & 31, nw = bs >> 5;
    for (int s = wid; s < SS; s += nw) {
        const float* rowp = &smem[s * MM];
        float d = 0.f, mm2 = 0.f;
        for (int m = lane; m < MM; m += 32) {
            const float mv = rowp[m];
            d   += mv * sh[m];
            mm2 += mv * mv;
        }
        d   = wave_sum(d);
        mm2 = wave_sum(mm2);
        if (lane == 0)
            ssim[s] = d / (fmaxf(sqrtf(mm2), 1e-6f) * hnorm);
    }
    __syncthreads();

    // ---- softmax over the S slots ----
    float lm = -3.402823466e38f;
    for (int s = tid; s < SS; s += bs) lm = fmaxf(lm, ssim[s]);
    const float mx = block_max(lm, sred);

    float ls = 0.f;
    for (int s = tid; s < SS; s += bs) {
        const float e = __expf(ssim[s] - mx);
        ssim[s] = e;                 // unnormalized weights
        ls += e;
    }
    const float sum = block_sum(ls, sred);   // barrier inside also publishes ssim[]
    const float scale = 1.f / (sum * (float)SS);

    // ---- pooled[m] = (sum_s e_s * mem[s][m]) * scale ;  g = pooled * h ----
    for (int m = tid; m < MM; m += bs) {
        float a = 0.f;
        for (int s = 0; s < SS; ++s)
            a += smem[s * MM + m] * ssim[s];
        g_ws[(size_t)n * MM + m] = a * scale * sh[m];
    }
}

// ---------------------------------------------------------------------------
extern "C" void kernel_launch(void* const* d_in, const int* in_sizes, int n_in,
                              void* d_out, int out_size, void* d_ws, size_t ws_size,
                              hipStream_t stream) {
    const float* x     = (const float*)d_in[0];
    const float* mem   = (const float*)d_in[1];
    const float* W_x   = (const float*)d_in[2];
    const float* b_x   = (const float*)d_in[3];
    const float* W_out = (const float*)d_in[4];
    const float* b_out = (const float*)d_in[5];
    float* out = (float*)d_out;

    float* h_ws = (float*)d_ws;                       // (N, M)
    float* g_ws = h_ws + (size_t)NN * MM;             // (N, M)

    // h = relu(x @ W_x + b_x)
    {
        const int tiles  = (NN / 16) * ((MM + 15) / 16);
        const int blocks = (tiles + 3) / 4;           // 4 waves / block
        gemm_bias_relu_wmma<<<blocks, 128, 0, stream>>>(
            x, W_x, b_x, h_ws, NN, XX, MM, XX, MM, MM);
    }

    // attention: g = (softmax(cos-sim) pooled mem / S) * h
    attn_kernel<<<NN, ATTN_BS, 0, stream>>>(mem, h_ws, g_ws);

    // out = relu(g @ W_out + b_out)
    {
        const int tiles  = (NN / 16) * ((OO + 15) / 16);
        const int blocks = (tiles + 3) / 4;
        gemm_bias_relu_wmma<<<blocks, 128, 0, stream>>>(
            g_ws, W_out, b_out, out, NN, MM, OO, MM, OO, OO);
    }
}